// Warp3d_60936995995709
// MI455X (gfx1250) — compile-verified
//
#include <hip/hip_runtime.h>

// 3D trilinear warp (voxelmorph grid_sample, align_corners=True, zero padding).
//   x  : (1, 3, 1, H, W, D) float32  -> 3 source volumes, contiguous NVOX floats each
//   w  : (1, 3, 3, H, W, D) float32  -> per-volume flow, component-major planes
//   out: (1, 3, 1, H, W, D) float32
//
// HBM-bound (~295 MB compulsory -> ~13 us floor @ 23.3 TB/s). This version:
//  - 2 consecutive k-voxels per thread: flow loads / output store become b64
//    vector ops (256 B per wave32 instruction), 16 independent gather taps in
//    flight per thread.
//  - Gather taps addressed as SGPR-base + 32-bit unsigned VGPR offset (volume
//    is 19.6 MB, offsets fit 32 bits) to avoid 64-bit VALU carry chains.
//  - NT temporal hints on streaming flow/output; speculative global_prefetch
//    of the likely gather line issued before the flow loads resolve.

#define VH 160
#define VW 192
#define VD 160
#define NVOX (VH * VW * VD)        // 4,915,200 voxels per volume
#define NTOT (3 * NVOX)            // 14,745,600 outputs
#define PAIRS (NTOT / 2)           // 7,372,800 threads

typedef float v2f __attribute__((ext_vector_type(2)));

__device__ __forceinline__ float trilerp_zeropad(const float* __restrict__ xv,
                                                 float cx, float cy, float cz) {
    const float x0f = floorf(cx);
    const float y0f = floorf(cy);
    const float z0f = floorf(cz);
    const float fx = cx - x0f;
    const float fy = cy - y0f;
    const float fz = cz - z0f;

    const int x0 = (int)x0f, x1 = x0 + 1;
    const int y0 = (int)y0f, y1 = y0 + 1;
    const int z0 = (int)z0f, z1 = z0 + 1;

    // Separable validity (zero padding) + clamped indices for the loads.
    const bool vx0 = (x0 >= 0) & (x0 < VH);
    const bool vx1 = (x1 >= 0) & (x1 < VH);
    const bool vy0 = (y0 >= 0) & (y0 < VW);
    const bool vy1 = (y1 >= 0) & (y1 < VW);
    const bool vz0 = (z0 >= 0) & (z0 < VD);
    const bool vz1 = (z1 >= 0) & (z1 < VD);

    const unsigned xc0 = (unsigned)min(max(x0, 0), VH - 1);
    const unsigned xc1 = (unsigned)min(max(x1, 0), VH - 1);
    const unsigned yc0 = (unsigned)min(max(y0, 0), VW - 1);
    const unsigned yc1 = (unsigned)min(max(y1, 0), VW - 1);
    const unsigned zc0 = (unsigned)min(max(z0, 0), VD - 1);
    const unsigned zc1 = (unsigned)min(max(z1, 0), VD - 1);

    // 32-bit row offsets off the uniform SGPR base pointer.
    const unsigned r00 = (xc0 * VW + yc0) * VD;
    const unsigned r01 = (xc0 * VW + yc1) * VD;
    const unsigned r10 = (xc1 * VW + yc0) * VD;
    const unsigned r11 = (xc1 * VW + yc1) * VD;

    // Predicated gathers (v_cndmask, no divergence); regular temporal hint —
    // these lines are reused heavily across neighboring threads/waves.
    const float v000 = (vx0 & vy0 & vz0) ? xv[r00 + zc0] : 0.0f;
    const float v001 = (vx0 & vy0 & vz1) ? xv[r00 + zc1] : 0.0f;
    const float v010 = (vx0 & vy1 & vz0) ? xv[r01 + zc0] : 0.0f;
    const float v011 = (vx0 & vy1 & vz1) ? xv[r01 + zc1] : 0.0f;
    const float v100 = (vx1 & vy0 & vz0) ? xv[r10 + zc0] : 0.0f;
    const float v101 = (vx1 & vy0 & vz1) ? xv[r10 + zc1] : 0.0f;
    const float v110 = (vx1 & vy1 & vz0) ? xv[r11 + zc0] : 0.0f;
    const float v111 = (vx1 & vy1 & vz1) ? xv[r11 + zc1] : 0.0f;

    // Fused hierarchical trilinear blend: 7 FMAs + 7 subs.
    const float c00 = fmaf(fz, v001 - v000, v000);
    const float c01 = fmaf(fz, v011 - v010, v010);
    const float c10 = fmaf(fz, v101 - v100, v100);
    const float c11 = fmaf(fz, v111 - v110, v110);
    const float c0  = fmaf(fy, c01 - c00, c00);
    const float c1  = fmaf(fy, c11 - c10, c10);
    return fmaf(fx, c1 - c0, c0);
}

__global__ __launch_bounds__(256) void warp3d_trilerp2_kernel(
    const float* __restrict__ x,   // (3, NVOX)
    const float* __restrict__ w,   // (3, 3, NVOX)
    float* __restrict__ out)       // (3, NVOX)
{
    const unsigned gid = blockIdx.x * 256u + threadIdx.x;
    if (gid >= (unsigned)PAIRS) return;

    const unsigned vv = gid * 2u;          // flat output index of first voxel
    const unsigned n  = vv / (unsigned)NVOX;   // volume 0..2
    const unsigned v  = vv - n * (unsigned)NVOX;
    const unsigned k  = v % (unsigned)VD;      // even; pair is (k, k+1), same row
    const unsigned t  = v / (unsigned)VD;
    const unsigned j  = t % (unsigned)VW;
    const unsigned i  = t / (unsigned)VW;

    const float* __restrict__ xv = x + (size_t)n * NVOX;
    const float* __restrict__ wv = w + (size_t)n * 3 * NVOX;

    // Flow ~N(0,1): both voxels' taps almost surely hit the cachelines around
    // (i,j,k..k+2). Issue the prefetch NOW, before the flow loads resolve,
    // to overlap the gather's HBM latency. gfx1250: global_prefetch_b8.
    __builtin_prefetch(xv + v, 0, 3);

    // Flow is streamed exactly once: b64 NT loads (256 B / wave instruction);
    // NT keeps the stream from evicting the gather-reused x volumes from L2.
    const v2f wx = __builtin_nontemporal_load((const v2f*)(wv + v));
    const v2f wy = __builtin_nontemporal_load((const v2f*)(wv + NVOX + v));
    const v2f wz = __builtin_nontemporal_load((const v2f*)(wv + 2 * NVOX + v));

    const float fi = (float)i;
    const float fj = (float)j;
    const float fk = (float)k;

    v2f res;
    res.x = trilerp_zeropad(xv, fi + wx.x, fj + wy.x, fk + wz.x);
    res.y = trilerp_zeropad(xv, fi + wx.y, fj + wy.y, fk + 1.0f + wz.y);

    // Write-once output: b64 NT store (write-rinse toward HBM, preserves L2
    // residency of the x volumes).
    __builtin_nontemporal_store(res, (v2f*)(out + (size_t)n * NVOX + v));
}

extern "C" void kernel_launch(void* const* d_in, const int* in_sizes, int n_in,
                              void* d_out, int out_size, void* d_ws, size_t ws_size,
                              hipStream_t stream) {
    const float* x = (const float*)d_in[0];   // (1,3,1,H,W,D)
    const float* w = (const float*)d_in[1];   // (1,3,3,H,W,D)
    float* out = (float*)d_out;               // (1,3,1,H,W,D)

    // PAIRS = 7,372,800 = 28,800 blocks x 256 threads (exact).
    dim3 block(256);
    dim3 grid((PAIRS + 255) / 256);
    warp3d_trilerp2_kernel<<<grid, block, 0, stream>>>(x, w, out);
}